// HH_8796093022490
// MI455X (gfx1250) — compile-verified
//
#include <hip/hip_runtime.h>

typedef float v2f __attribute__((ext_vector_type(2)));
typedef float v8f __attribute__((ext_vector_type(8)));

#define SDIM 512
#define NREF 514
#define HH_EPS 1e-16f

// ---------------------------------------------------------------------------
// Kernel 1: coefs[i] = 2 / (v_i . v_i + eps).  One wave32 per reflector.
// Lane l owns elements [l*16, l*16+16); xor-shuffle reduction across the wave.
// ---------------------------------------------------------------------------
__global__ void hh_coef_kernel(const float* __restrict__ v,
                               float* __restrict__ coefs) {
    int wid  = (blockIdx.x * blockDim.x + threadIdx.x) >> 5;
    int lane = threadIdx.x & 31;
    if (wid >= NREF) return;
    const float4* vp = (const float4*)(v + wid * SDIM + lane * 16);
    float s = 0.0f;
#pragma unroll
    for (int q = 0; q < 4; ++q) {
        float4 t = vp[q];
        s += t.x * t.x + t.y * t.y + t.z * t.z + t.w * t.w;
    }
#pragma unroll
    for (int off = 16; off >= 1; off >>= 1) s += __shfl_xor(s, off, 32);
    if (lane == 0) coefs[wid] = 2.0f / (s + HH_EPS);
}

// ---------------------------------------------------------------------------
// Kernel 2: build W = Q^T = M_{513} ... M_0 one column per wave32.
// Column j starts as e_j; apply c -= coef_i * v_i * (v_i . c) for i=0..513.
// Lane l holds c[l*16 + t], t=0..15.  Result stored COLUMN-MAJOR:
// Wc[j*512 + k] = W[k][j].  v_i (2KB) stays hot in WGP$/L2.
// ---------------------------------------------------------------------------
__global__ void hh_col_kernel(const float* __restrict__ v,
                              const float* __restrict__ coefs,
                              float* __restrict__ Wc) {
    int wid  = threadIdx.x >> 5;
    int lane = threadIdx.x & 31;
    int col  = blockIdx.x * (blockDim.x >> 5) + wid;   // 0..511

    float c[16];
#pragma unroll
    for (int t = 0; t < 16; ++t)
        c[t] = ((lane * 16 + t) == col) ? 1.0f : 0.0f;

    for (int i = 0; i < NREF; ++i) {
        const float4* vp = (const float4*)(v + i * SDIM + lane * 16);
        float vr[16];
#pragma unroll
        for (int q = 0; q < 4; ++q) {
            float4 t = vp[q];
            vr[q * 4 + 0] = t.x; vr[q * 4 + 1] = t.y;
            vr[q * 4 + 2] = t.z; vr[q * 4 + 3] = t.w;
        }
        float dot = 0.0f;
#pragma unroll
        for (int t = 0; t < 16; ++t) dot = fmaf(vr[t], c[t], dot);
#pragma unroll
        for (int off = 16; off >= 1; off >>= 1) dot += __shfl_xor(dot, off, 32);
        float a = coefs[i] * dot;
#pragma unroll
        for (int t = 0; t < 16; ++t) c[t] = fmaf(-a, vr[t], c[t]);
    }

    float4* out = (float4*)(Wc + col * SDIM + lane * 16);
#pragma unroll
    for (int q = 0; q < 4; ++q) {
        float4 t;
        t.x = c[q * 4 + 0]; t.y = c[q * 4 + 1];
        t.z = c[q * 4 + 2]; t.w = c[q * 4 + 3];
        out[q] = t;
    }
}

// ---------------------------------------------------------------------------
// Kernel 3: y[65536,512] = x[65536,512] @ W  via V_WMMA_F32_16X16X4_F32.
// One wave computes a 16(M) x 128(N) strip: 8 accumulators (v8f each),
// A fragment reused across the 8 N-tiles per K-step.
// A layout (16x4 f32): lane<16 -> x[m0+l][k0 + {0,1}], lane>=16 -> K=2,3.
// B layout mirrors it; with Wc column-major, both are float2 loads.
// ---------------------------------------------------------------------------
__global__ void __launch_bounds__(256) hh_gemm_kernel(
        const float* __restrict__ x,
        const float* __restrict__ Wc,
        float* __restrict__ y) {
    int wid  = threadIdx.x >> 5;
    int lane = threadIdx.x & 31;
    int gw    = blockIdx.x * 8 + wid;
    int mTile = gw >> 2;          // 0..4095 : 16 rows of x each
    int nSlab = gw & 3;           // 0..3    : 128 columns of W each
    int half  = lane >> 4;        // K sub-phase (0: K=0,1 ; 1: K=2,3)
    int l15   = lane & 15;

    const float* ap    = x  + (mTile * 16  + l15) * SDIM + 2 * half;
    const float* bbase = Wc + (nSlab * 128 + l15) * SDIM + 2 * half;

    v8f acc[8] = {};

    for (int k0 = 0; k0 < SDIM; k0 += 4) {
        v2f a = *(const v2f*)(ap + k0);
#pragma unroll
        for (int nt = 0; nt < 8; ++nt) {
            v2f b = *(const v2f*)(bbase + nt * 16 * SDIM + k0);
            acc[nt] = __builtin_amdgcn_wmma_f32_16x16x4_f32(
                false, a, false, b, (short)0, acc[nt], false, false);
        }
    }

    // C/D layout: VGPR e -> (lanes 0-15: M=e, N=l15) (lanes 16-31: M=e+8)
#pragma unroll
    for (int nt = 0; nt < 8; ++nt) {
        int ncol = nSlab * 128 + nt * 16 + l15;
#pragma unroll
        for (int e = 0; e < 8; ++e) {
            y[(mTile * 16 + e + 8 * half) * SDIM + ncol] = acc[nt][e];
        }
    }
}

// ---------------------------------------------------------------------------
extern "C" void kernel_launch(void* const* d_in, const int* in_sizes, int n_in,
                              void* d_out, int out_size, void* d_ws, size_t ws_size,
                              hipStream_t stream) {
    const float* x = (const float*)d_in[0];   // [65536, 512]
    const float* v = (const float*)d_in[1];   // [514, 512, 1]
    float* y  = (float*)d_out;                // [65536, 512]

    float* Wc    = (float*)d_ws;              // 512*512 floats, column-major W
    float* coefs = Wc + SDIM * SDIM;          // 514 floats

    // 514 waves -> 65 blocks of 8 waves
    hh_coef_kernel<<<(NREF * 32 + 255) / 256, 256, 0, stream>>>(v, coefs);

    // 512 columns, 8 waves (columns) per block -> 64 blocks
    hh_col_kernel<<<SDIM / 8, 256, 0, stream>>>(v, coefs, Wc);

    // 4096 mTiles * 4 nSlabs = 16384 waves -> 2048 blocks of 8 waves
    hh_gemm_kernel<<<2048, 256, 0, stream>>>(x, Wc, y);
}